// GNNML1_2181843387146
// MI455X (gfx1250) — compile-verified
//
#include <hip/hip_runtime.h>
#include <hip/hip_bf16.h>

typedef __attribute__((ext_vector_type(2))) float v2f;
typedef __attribute__((ext_vector_type(8))) float v8f;

#define N_NODES 100000
#define N_EDGES 1600000
#define FEAT 64
#define NUM_GRAPHS 256

// ---------------------------------------------------------------- zero fill
__global__ __launch_bounds__(256) void zero_kernel(float* __restrict__ p, int n) {
    int i = blockIdx.x * 256 + threadIdx.x;
    if (i < n) p[i] = 0.0f;
}

// ------------------------------------------------- edge scatter: agg[dst] += H[src]
// 16 threads per edge, 4 features each (float4 gather, 4 scalar f32 atomics).
__global__ __launch_bounds__(256) void scatter_kernel(
    const float* __restrict__ H, const int* __restrict__ src,
    const int* __restrict__ dst, float* __restrict__ agg, int n_edges)
{
    int gid = blockIdx.x * 256 + threadIdx.x;
    int e = gid >> 4;
    if (e >= n_edges) return;
    int f = (gid & 15) * 4;
    int s = src[e];
    int d = dst[e];
    const float4 v = *(const float4*)(H + (size_t)s * FEAT + f);
    float* a = agg + (size_t)d * FEAT + f;
    atomicAdd(a + 0, v.x);
    atomicAdd(a + 1, v.y);
    atomicAdd(a + 2, v.z);
    atomicAdd(a + 3, v.w);
}

// ------------------------------------------------- fused GNN layer (WMMA f32)
// Hout = relu( Hin@Wa + Ba  +  Agg@Wc + Bc  +  (Hin@Wm1 + Bm1) * (Hin@Wm2 + Bm2) )
// One wave32 computes a 16-row x 64-col output tile. K=64 via 16 steps of
// V_WMMA_F32_16X16X4_F32. Weights (4 x 64x64 f32 = 64KB) staged in LDS.
__global__ __launch_bounds__(256) void fused_layer_kernel(
    const float* __restrict__ Hin, const float* __restrict__ Agg,
    const float* __restrict__ Wa,  const float* __restrict__ Ba,
    const float* __restrict__ Wc,  const float* __restrict__ Bc,
    const float* __restrict__ Wm1, const float* __restrict__ Bm1,
    const float* __restrict__ Wm2, const float* __restrict__ Bm2,
    float* __restrict__ Hout, int n_nodes)
{
    __shared__ float lW[4 * FEAT * FEAT];   // 64 KB: Wa | Wc | Wm1 | Wm2

    const int t = threadIdx.x;
    for (int i = t; i < FEAT * FEAT; i += 256) {
        lW[0 * 4096 + i] = Wa[i];
        lW[1 * 4096 + i] = Wc[i];
        lW[2 * 4096 + i] = Wm1[i];
        lW[3 * 4096 + i] = Wm2[i];
    }
    __syncthreads();

    const int wave  = t >> 5;                       // 0..7 (wave32)
    const int lane  = t & 31;
    const int tile  = blockIdx.x * 8 + wave;        // 16-row tile index
    if (tile * 16 >= n_nodes) return;

    const int r0    = tile * 16;
    const int lrow  = lane & 15;                    // M (A) / N (B,C,D)
    const int khalf = lane >> 4;                    // 0: K={0,1}, 1: K={2,3}

    const float* hrow = Hin + (size_t)(r0 + lrow) * FEAT;
    const float* grow = Agg + (size_t)(r0 + lrow) * FEAT;

    for (int nt = 0; nt < 4; ++nt) {
        const int n0   = nt * 16;
        const int ncol = n0 + lrow;

        v8f accA = {}; v8f accC = {}; v8f accM1 = {}; v8f accM2 = {};

        for (int kb = 0; kb < FEAT; kb += 4) {
            const int kk = kb + khalf * 2;

            v2f aH; aH.x = hrow[kk]; aH.y = hrow[kk + 1];
            v2f aG; aG.x = grow[kk]; aG.y = grow[kk + 1];

            v2f bA;  bA.x  = lW[0 * 4096 + kk * FEAT + ncol];
                     bA.y  = lW[0 * 4096 + (kk + 1) * FEAT + ncol];
            v2f bC;  bC.x  = lW[1 * 4096 + kk * FEAT + ncol];
                     bC.y  = lW[1 * 4096 + (kk + 1) * FEAT + ncol];
            v2f bM1; bM1.x = lW[2 * 4096 + kk * FEAT + ncol];
                     bM1.y = lW[2 * 4096 + (kk + 1) * FEAT + ncol];
            v2f bM2; bM2.x = lW[3 * 4096 + kk * FEAT + ncol];
                     bM2.y = lW[3 * 4096 + (kk + 1) * FEAT + ncol];

            accA  = __builtin_amdgcn_wmma_f32_16x16x4_f32(false, aH, false, bA,  (short)0, accA,  false, false);
            accC  = __builtin_amdgcn_wmma_f32_16x16x4_f32(false, aG, false, bC,  (short)0, accC,  false, false);
            accM1 = __builtin_amdgcn_wmma_f32_16x16x4_f32(false, aH, false, bM1, (short)0, accM1, false, false);
            accM2 = __builtin_amdgcn_wmma_f32_16x16x4_f32(false, aH, false, bM2, (short)0, accM2, false, false);
        }

        const float ba  = Ba[ncol];
        const float bc  = Bc[ncol];
        const float bm1 = Bm1[ncol];
        const float bm2 = Bm2[ncol];

        #pragma unroll
        for (int i = 0; i < 8; ++i) {
            float v = (accA[i] + ba) + (accC[i] + bc)
                    + (accM1[i] + bm1) * (accM2[i] + bm2);
            v = v > 0.0f ? v : 0.0f;
            Hout[(size_t)(r0 + i + 8 * khalf) * FEAT + ncol] = v;
        }
    }
}

// ------------------------------------------------- mean-pool accumulation
__global__ __launch_bounds__(256) void pool_accum(
    const float* __restrict__ H, const int* __restrict__ batch,
    float* __restrict__ sums, float* __restrict__ cnts, int n_nodes)
{
    int gid = blockIdx.x * 256 + threadIdx.x;
    int node = gid >> 6;
    if (node >= n_nodes) return;
    int f = gid & 63;
    int g = batch[node];
    atomicAdd(&sums[g * FEAT + f], H[(size_t)node * FEAT + f]);
    if (f == 0) atomicAdd(&cnts[g], 1.0f);
}

// ------------------------------------------------- head: mean, 64->10->1
__global__ __launch_bounds__(256) void head_kernel(
    const float* __restrict__ sums, const float* __restrict__ cnts,
    const float* __restrict__ fc1w, const float* __restrict__ fc1b,
    const float* __restrict__ fc2w, const float* __restrict__ fc2b,
    float* __restrict__ out)
{
    int g = threadIdx.x;
    if (g >= NUM_GRAPHS) return;
    float c = cnts[g];
    c = c > 1.0f ? c : 1.0f;
    float pooled[FEAT];
    #pragma unroll
    for (int f = 0; f < FEAT; ++f) pooled[f] = sums[g * FEAT + f] / c;
    float r = fc2b[0];
    for (int j = 0; j < 10; ++j) {
        float a = fc1b[j];
        for (int f = 0; f < FEAT; ++f) a += pooled[f] * fc1w[f * 10 + j];
        r += a * fc2w[j];
    }
    out[g] = r;
}

// ---------------------------------------------------------------- launcher
extern "C" void kernel_launch(void* const* d_in, const int* in_sizes, int n_in,
                              void* d_out, int out_size, void* d_ws, size_t ws_size,
                              hipStream_t stream) {
    const float* x     = (const float*)d_in[0];
    const int*   eidx  = (const int*)d_in[1];
    const int*   batch = (const int*)d_in[2];

    const int n_nodes = in_sizes[0] / FEAT;
    const int n_edges = in_sizes[1] / 2;
    const int* src = eidx;
    const int* dst = eidx + n_edges;

    // per-layer param indices: conv_w, conv_b, fc{l}1_w, fc{l}1_b, fc{l}2_w, fc{l}2_b, fc{l}3_w, fc{l}3_b
    // layer l base = 3 + (l-1)*8
    const float* fc1w = (const float*)d_in[27];
    const float* fc1b = (const float*)d_in[28];
    const float* fc2w = (const float*)d_in[29];
    const float* fc2b = (const float*)d_in[30];

    // workspace layout (floats)
    float* ws   = (float*)d_ws;
    const size_t NF = (size_t)n_nodes * FEAT;       // 6.4M floats
    float* buf0 = ws;
    float* buf1 = ws + NF;
    float* agg  = ws + 2 * NF;
    float* sums = ws + 3 * NF;                      // 256*64
    float* cnts = sums + NUM_GRAPHS * FEAT;         // 256

    const int zeroAggBlocks = (int)((NF + 255) / 256);
    const int scatBlocks    = (n_edges * 16 + 255) / 256;
    const int tileBlocks    = ((n_nodes + 15) / 16 + 7) / 8;
    const int poolBlocks    = (n_nodes * FEAT + 255) / 256;

    const float* layer_in[3]  = { x, buf0, buf1 };
    float*       layer_out[3] = { buf0, buf1, buf0 };

    for (int l = 0; l < 3; ++l) {
        const int base = 3 + l * 8;
        const float* cw  = (const float*)d_in[base + 0];
        const float* cb  = (const float*)d_in[base + 1];
        const float* faw = (const float*)d_in[base + 2];
        const float* fab = (const float*)d_in[base + 3];
        const float* fbw = (const float*)d_in[base + 4];
        const float* fbb = (const float*)d_in[base + 5];
        const float* fcw = (const float*)d_in[base + 6];
        const float* fcb = (const float*)d_in[base + 7];

        zero_kernel<<<zeroAggBlocks, 256, 0, stream>>>(agg, (int)NF);
        scatter_kernel<<<scatBlocks, 256, 0, stream>>>(layer_in[l], src, dst, agg, n_edges);
        fused_layer_kernel<<<tileBlocks, 256, 0, stream>>>(
            layer_in[l], agg, faw, fab, cw, cb, fbw, fbb, fcw, fcb,
            layer_out[l], n_nodes);
    }

    zero_kernel<<<(NUM_GRAPHS * FEAT + NUM_GRAPHS + 255) / 256, 256, 0, stream>>>(
        sums, NUM_GRAPHS * FEAT + NUM_GRAPHS);
    pool_accum<<<poolBlocks, 256, 0, stream>>>(layer_out[2], batch, sums, cnts, n_nodes);
    head_kernel<<<1, 256, 0, stream>>>(sums, cnts, fc1w, fc1b, fc2w, fc2b, (float*)d_out);
}